// Attention_59450937311976
// MI455X (gfx1250) — compile-verified
//
#include <hip/hip_runtime.h>
#include <math.h>

typedef float v2f __attribute__((ext_vector_type(2)));
typedef float v8f __attribute__((ext_vector_type(8)));

namespace {
constexpr int Bsz = 16, TRG = 512, SRC = 1024, ENC = 1024, TRGD = 1024;
}

// C[M,N] = A[M,K] @ op(B), all row-major.
//   BT = true : op(B) = B^T with B stored as [N,K] (contract fast dim of B)
//   BT = false: op(B) = B   with B stored as [K,N]
// MASK_N: multiply column n of C by maskBase[b*N + n] at store time.
// Wave tile 32x64 (2x4 WMMA 16x16 tiles), 8 waves -> 64x256 per workgroup.
template <bool BT, bool MASK_N>
__global__ __launch_bounds__(256) void gemm_wmma_f32(
    const float* __restrict__ Abase, const float* __restrict__ Bbase,
    float* __restrict__ Cbase, const float* __restrict__ maskBase,
    int M, int N, int K, long long sA, long long sB, long long sC) {
  const int b = blockIdx.z;
  const float* A = Abase + (long long)b * sA;
  const float* Bm = Bbase + (long long)b * sB;
  float* C = Cbase + (long long)b * sC;

  const int lane = threadIdx.x & 31;
  const int wave = threadIdx.x >> 5;

  const int m0 = blockIdx.x * 64 + (wave >> 2) * 32;   // wave rows
  const int n0 = blockIdx.y * 256 + (wave & 3) * 64;   // wave cols

  const int r16 = lane & 15;   // M row (A) / N col (B,C) within tile
  const int kh = lane >> 4;    // K half-select (0 -> k0,k1 ; 1 -> k2,k3)

  v8f acc[2][4] = {};

  // A: lane holds float2 {A[m][k+2*kh], A[m][k+2*kh+1]}
  const float* aPtr0 = A + (long long)(m0 + r16) * K + 2 * kh;
  const float* aPtr1 = aPtr0 + 16LL * K;

#pragma unroll 4
  for (int k = 0; k < K; k += 4) {
    v2f a0 = *(const v2f*)(aPtr0 + k);
    v2f a1 = *(const v2f*)(aPtr1 + k);

    v2f bt[4];
    if (BT) {
      // B stored [N,K]: lane n reads float2 at B[n][k+2*kh] (mirror of A layout)
#pragma unroll
      for (int nt = 0; nt < 4; ++nt)
        bt[nt] = *(const v2f*)(Bm + (long long)(n0 + nt * 16 + r16) * K + k + 2 * kh);
    } else {
      // B stored [K,N]: lane n reads B[k+2*kh][n] and B[k+2*kh+1][n]
      const float* brow = Bm + (long long)(k + 2 * kh) * N + n0 + r16;
#pragma unroll
      for (int nt = 0; nt < 4; ++nt) {
        v2f v;
        v.x = brow[nt * 16];
        v.y = brow[nt * 16 + N];
        bt[nt] = v;
      }
    }

#pragma unroll
    for (int nt = 0; nt < 4; ++nt) {
      acc[0][nt] = __builtin_amdgcn_wmma_f32_16x16x4_f32(
          false, a0, false, bt[nt], (short)0, acc[0][nt], false, false);
      acc[1][nt] = __builtin_amdgcn_wmma_f32_16x16x4_f32(
          false, a1, false, bt[nt], (short)0, acc[1][nt], false, false);
    }
  }

  // C/D layout: VGPR r, lanes 0-15 -> M = r, N = lane; lanes 16-31 -> M = r+8.
#pragma unroll
  for (int mt = 0; mt < 2; ++mt) {
#pragma unroll
    for (int nt = 0; nt < 4; ++nt) {
      const int cn = n0 + nt * 16 + r16;
      float mv = 1.0f;
      if (MASK_N) mv = maskBase[(long long)b * N + cn];
      float* cp = C + (long long)(m0 + mt * 16 + 8 * kh) * N + cn;
#pragma unroll
      for (int r = 0; r < 8; ++r) cp[(long long)r * N] = acc[mt][nt][r] * mv;
    }
  }
}

// One wave per (b,t) row of S=1024 elements (held in registers as float4).
// x is the pre-masked energies (zeros at masked positions), matching the
// reference's x = attn_energies*mask since mask in {0,1}.
__global__ __launch_bounds__(256) void masked_softmax_rows(
    const float* __restrict__ x, const float* __restrict__ mask,
    float* __restrict__ w, int T, int S) {
  const int lane = threadIdx.x & 31;
  const int row = blockIdx.x * (blockDim.x >> 5) + (threadIdx.x >> 5);
  const int b = row / T;

  const float4* xr = (const float4*)(x + (long long)row * S);
  const float4* mr = (const float4*)(mask + (long long)b * S);

  float4 xv[8], mv[8];
  float mx = -3.402823466e38f;
#pragma unroll
  for (int i = 0; i < 8; ++i) {
    float4 v = xr[i * 32 + lane];
    float4 m = mr[i * 32 + lane];
    xv[i] = v;
    mv[i] = m;
    mx = fmaxf(mx, fmaxf(fmaxf(v.x, v.y), fmaxf(v.z, v.w)));
  }
#pragma unroll
  for (int o = 16; o; o >>= 1) mx = fmaxf(mx, __shfl_xor(mx, o, 32));

  float s = 0.0f;
#pragma unroll
  for (int i = 0; i < 8; ++i) {
    float4 e;
    e.x = __expf(xv[i].x - mx) * mv[i].x;
    e.y = __expf(xv[i].y - mx) * mv[i].y;
    e.z = __expf(xv[i].z - mx) * mv[i].z;
    e.w = __expf(xv[i].w - mx) * mv[i].w;
    xv[i] = e;
    s += (e.x + e.y) + (e.z + e.w);
  }
#pragma unroll
  for (int o = 16; o; o >>= 1) s += __shfl_xor(s, o, 32);

  const float inv = 1.0f / (s + 1e-6f);
  float4* wr = (float4*)(w + (long long)row * S);
#pragma unroll
  for (int i = 0; i < 8; ++i) {
    float4 o;
    o.x = xv[i].x * inv;
    o.y = xv[i].y * inv;
    o.z = xv[i].z * inv;
    o.w = xv[i].w * inv;
    wr[i * 32 + lane] = o;
  }
}

extern "C" void kernel_launch(void* const* d_in, const int* in_sizes, int n_in,
                              void* d_out, int out_size, void* d_ws,
                              size_t ws_size, hipStream_t stream) {
  (void)in_sizes; (void)n_in; (void)out_size; (void)d_ws; (void)ws_size;

  const float* hidden = (const float*)d_in[0];   // (B, T, D)
  const float* enc_out = (const float*)d_in[1];  // (B, S, E)
  const float* enc_val = (const float*)d_in[2];  // (B, S, D)
  const float* mask = (const float*)d_in[3];     // (B, S)
  const float* W = (const float*)d_in[4];        // (D, E)

  float* out = (float*)d_out;
  float* ctx = out;                                    // (B, T, D)
  float* attw = out + (size_t)Bsz * TRG * TRGD;        // (B, T, S)
  float* atte = attw + (size_t)Bsz * TRG * SRC;        // (B, T, S)

  const dim3 blk(256);

  // 1) hW[b] = hidden[b] @ W   (T x E), staged in the context region.
  gemm_wmma_f32<false, false><<<dim3(TRG / 64, ENC / 256, Bsz), blk, 0, stream>>>(
      hidden, W, ctx, nullptr, TRG, ENC, TRGD,
      (long long)TRG * TRGD, 0LL, (long long)TRG * ENC);

  // 2) attn_energies[b] = (hW[b] @ enc_out[b]^T) * mask[b,s]   (T x S)
  gemm_wmma_f32<true, true><<<dim3(TRG / 64, SRC / 256, Bsz), blk, 0, stream>>>(
      ctx, enc_out, atte, mask, TRG, SRC, ENC,
      (long long)TRG * ENC, (long long)SRC * ENC, (long long)TRG * SRC);

  // 3) masked softmax over s -> attn_weights
  masked_softmax_rows<<<dim3(Bsz * TRG / 8), blk, 0, stream>>>(
      atte, mask, attw, TRG, SRC);

  // 4) context[b] = attn_weights[b] @ enc_val[b]   (T x D), overwrites hW.
  gemm_wmma_f32<false, false><<<dim3(TRG / 64, TRGD / 256, Bsz), blk, 0, stream>>>(
      attw, enc_val, ctx, nullptr, TRG, TRGD, SRC,
      (long long)TRG * SRC, (long long)SRC * TRGD, (long long)TRG * TRGD);
}